// ProperNeuronLLM_Fast_66391604462403
// MI455X (gfx1250) — compile-verified
//
#include <hip/hip_runtime.h>
#include <hip/hip_bf16.h>
#include <math.h>

#define SEQ_LEN 48
#define VOCAB   16000
#define EMBED   128
#define BATCH   16
#define EPS     1e-8f
#define SCALE   10.0f
#define NTILES_PER_WAVE 8   // 1000 tiles / 8 = 125 chunks exactly

typedef __attribute__((ext_vector_type(2))) float v2f;
typedef __attribute__((ext_vector_type(8))) float v8f;

// ---------------------------------------------------------------------------
// Stage 1: gather + exclusive causal prefix-sum + L2 normalize.
// Output layout ws_ctxn[s][b][d]  (S x 16 x 128 f32, 384 KB) chosen so the
// WMMA A-fragment (16x4 f32: lanes 0-15 hold K={k,k+1}, lanes 16-31 K={k+2,k+3})
// is a single b64 load per k-step.
// ---------------------------------------------------------------------------
__global__ __launch_bounds__(EMBED)
void ctx_prefix_kernel(const int*   __restrict__ tokens,   // [B][S]
                       const float* __restrict__ scalars,  // [S][V]
                       const float* __restrict__ wvec,     // [S][V][D]
                       float*       __restrict__ ctxn)     // [S][16][D]
{
    const int b = blockIdx.x;      // 0..15
    const int d = threadIdx.x;     // 0..127
    __shared__ float red[EMBED];

    float running = 0.0f;          // sum_{s' < s} act*w  (exclusive)
    for (int s = 0; s < SEQ_LEN; ++s) {
        // ||running||^2 via block reduction
        red[d] = running * running;
        __syncthreads();
        for (int off = EMBED / 2; off > 0; off >>= 1) {
            if (d < off) red[d] += red[d + off];
            __syncthreads();
        }
        const float inv = 1.0f / fmaxf(sqrtf(red[0]), EPS);
        __syncthreads();   // red[] reused next iteration

        ctxn[((size_t)s * BATCH + b) * EMBED + d] = running * inv;

        const int   tok = tokens[b * SEQ_LEN + s];
        const float a   = scalars[(size_t)s * VOCAB + tok];
        const float w   = wvec[((size_t)s * VOCAB + tok) * EMBED + d];
        running = fmaf(a, w, running);
    }
}

// ---------------------------------------------------------------------------
// Stage 2: per-position GEMM  [16 x 128] x [128 x 16000] with fused
// embedding-row normalization.  One wave per 8 output tiles of 16x16.
//   A (M=16 batch, K=128)  : preloaded from ws_ctxn, reused for all tiles
//   B (K=128,  N=16 vocab) : emb^T fragments, b64 loads; squares accumulated
//                            in-register to build per-column inverse norms.
// ---------------------------------------------------------------------------
__global__ __launch_bounds__(32)
void logits_wmma_kernel(const float* __restrict__ ctxn, // [S][16][128]
                        const float* __restrict__ emb,  // [S][V][128]
                        float*       __restrict__ out)  // [B][S][V]
{
    const int s       = blockIdx.y;          // 0..47
    const int chunk   = blockIdx.x;          // 0..124
    const int lane    = threadIdx.x;         // 0..31 (wave32)
    const int halfsel = (lane >> 4) & 1;     // K sub-pair select
    const int lm      = lane & 15;           // M (A) / N (B,C) index

    // Preload the full A panel: 32 k-steps of 16x4 fragments (64 VGPRs).
    v2f a[32];
    const float* abase = ctxn + ((size_t)s * BATCH + lm) * EMBED + 2 * halfsel;
    #pragma unroll
    for (int k = 0; k < 32; ++k)
        a[k] = *(const v2f*)(abase + 4 * k);

    const size_t emb_s = (size_t)s * VOCAB;

    for (int t = 0; t < NTILES_PER_WAVE; ++t) {
        const int v0 = (chunk * NTILES_PER_WAVE + t) * 16;
        const float* bbase = emb + (emb_s + v0 + lm) * EMBED + 2 * halfsel;

        // Prefetch next tile's B stream (emits global_prefetch_b8).
        if (t + 1 < NTILES_PER_WAVE)
            __builtin_prefetch(bbase + 16 * EMBED, 0, 1);

        v8f   c  = {};
        float sq = 0.0f;   // partial sum of squares of column (v0+lm)
        #pragma unroll
        for (int k = 0; k < 32; ++k) {
            v2f bf = *(const v2f*)(bbase + 4 * k);
            sq = fmaf(bf.x, bf.x, sq);
            sq = fmaf(bf.y, bf.y, sq);
            // D = A x B + C  : v_wmma_f32_16x16x4_f32
            c = __builtin_amdgcn_wmma_f32_16x16x4_f32(
                    /*neg_a=*/false, a[k],
                    /*neg_b=*/false, bf,
                    /*c_mod=*/(short)0, c,
                    /*reuse_a=*/false, /*reuse_b=*/false);
        }

        // lanes n and n+16 jointly hold column n's squares -> combine.
        sq += __shfl_xor(sq, 16, 32);
        const float colscale = SCALE / fmaxf(sqrtf(sq), EPS);

        // C layout: VGPR r -> lanes 0-15: (M=r, N=lane), lanes 16-31: (M=r+8).
        #pragma unroll
        for (int r = 0; r < 8; ++r) {
            const int brow = r + 8 * halfsel;
            out[((size_t)brow * SEQ_LEN + s) * VOCAB + v0 + lm] = c[r] * colscale;
        }
    }
}

// ---------------------------------------------------------------------------
extern "C" void kernel_launch(void* const* d_in, const int* in_sizes, int n_in,
                              void* d_out, int out_size, void* d_ws, size_t ws_size,
                              hipStream_t stream) {
    const int*   tokens  = (const int*)  d_in[0];   // [16][48]   int32
    const float* scalars = (const float*)d_in[1];   // [48][16000] f32
    const float* wvec    = (const float*)d_in[2];   // [48][16000][128] f32
    const float* emb     = (const float*)d_in[3];   // [48][16000][128] f32
    float*       logits  = (float*)d_out;           // [16][48][16000] f32
    float*       ctxn    = (float*)d_ws;            // [48][16][128] f32 (384 KB)

    ctx_prefix_kernel<<<BATCH, EMBED, 0, stream>>>(tokens, scalars, wvec, ctxn);

    dim3 grid(VOCAB / (16 * NTILES_PER_WAVE), SEQ_LEN);  // (125, 48)
    logits_wmma_kernel<<<grid, 32, 0, stream>>>(ctxn, emb, logits);
}